// ReactionOptimizerGNN_15685220565725
// MI455X (gfx1250) — compile-verified
//
#include <hip/hip_runtime.h>
#include <hip/hip_bf16.h>

typedef float v2f __attribute__((ext_vector_type(2)));
typedef float v8f __attribute__((ext_vector_type(8)));

static __device__ __forceinline__ void atom_add_f32(float* p, float v) {
  // relaxed, agent-scope fp32 atomic add -> global_atomic_add_f32 (no return)
  __hip_atomic_fetch_add(p, v, __ATOMIC_RELAXED, __HIP_MEMORY_SCOPE_AGENT);
}

__global__ void k_zero(float* __restrict__ p, int n) {
  int i = blockIdx.x * blockDim.x + threadIdx.x;
  if (i < n) p[i] = 0.0f;
}

__global__ void k_deg(const int* __restrict__ dst, float* __restrict__ deg, int E) {
  int i = blockIdx.x * blockDim.x + threadIdx.x;
  if (i < E) atom_add_f32(&deg[dst[i]], 1.0f);
}

__global__ void k_dinv(const float* __restrict__ deg, float* __restrict__ dinv, int n) {
  int i = blockIdx.x * blockDim.x + threadIdx.x;
  if (i < n) dinv[i] = rsqrtf(deg[i] + 1.0f);
}

__global__ void k_edgenorm(const int* __restrict__ src, const int* __restrict__ dst,
                           const float* __restrict__ dinv, float* __restrict__ en, int E) {
  int i = blockIdx.x * blockDim.x + threadIdx.x;
  if (i < E) en[i] = dinv[src[i]] * dinv[dst[i]];
}

// ---------------------------------------------------------------------------
// GEMM: out[N,M] = (RELU_IN ? relu(A) : A)[N,K] @ W[K,M] using
// V_WMMA_F32_16X16X4_F32. One wave per 16x16 output tile; a block covers a
// 16-row strip (M/16 waves). A-strip staged in LDS (stride K+4 -> conflict-free
// ds_load_b64 for the A fragment).
// A frag (16x4 f32): lane%16 = M row, VGPR j holds K = k + j + 2*(lane>=16).
// B frag (4x16 f32): lane%16 = N col, VGPR j holds K = k + j + 2*(lane>=16).
// C/D (16x16 f32):   VGPR r = row r + 8*(lane>=16), col = lane%16.
// ---------------------------------------------------------------------------
template <int K, int M, bool RELU_IN>
__global__ __launch_bounds__(M * 2) void k_gemm_wmma(const float* __restrict__ A,
                                                     const float* __restrict__ W,
                                                     float* __restrict__ out, int N) {
  __shared__ float As[16][K + 4];
  const int r0 = blockIdx.x * 16;
  const int tid = threadIdx.x;

  for (int idx = tid; idx < 16 * K; idx += M * 2) {
    int r = idx / K;
    int c = idx - r * K;
    float v = 0.0f;
    int row = r0 + r;
    if (row < N) v = A[(size_t)row * K + c];
    if (RELU_IN) v = fmaxf(v, 0.0f);
    As[r][c] = v;
  }
  __syncthreads();

  const int lane = tid & 31;
  const int wv   = tid >> 5;
  const int col  = lane & 15;   // M row (for A) / N col (for B, D)
  const int hi   = lane >> 4;   // upper half-wave holds K+2..K+3
  const int n0   = wv * 16;

  v8f acc = {0.0f, 0.0f, 0.0f, 0.0f, 0.0f, 0.0f, 0.0f, 0.0f};

#pragma unroll
  for (int k = 0; k < K; k += 4) {
    const int kk = k + 2 * hi;
    v2f a = *(const v2f*)&As[col][kk];            // 8B-aligned (kk even, stride K+4)
    v2f b;
    b.x = W[(size_t)kk * M + n0 + col];
    b.y = W[(size_t)(kk + 1) * M + n0 + col];
    acc = __builtin_amdgcn_wmma_f32_16x16x4_f32(
        /*neg_a=*/false, a, /*neg_b=*/false, b,
        /*c_mod=*/(short)0, acc, /*reuse_a=*/false, /*reuse_b=*/false);
  }

#pragma unroll
  for (int r = 0; r < 8; ++r) {
    int row = r0 + r + 8 * hi;
    if (row < N) out[(size_t)row * M + n0 + col] = acc[r];
  }
}

// agg[i,c] = t[i,c] * dinv[i]^2 + b[c]   (self-loop contribution + bias)
template <int M>
__global__ void k_self(const float* __restrict__ t, const float* __restrict__ dinv,
                       const float* __restrict__ b, float* __restrict__ agg, int N) {
  long long i = (long long)blockIdx.x * blockDim.x + threadIdx.x;
  long long total = (long long)N * M;
  if (i >= total) return;
  int node = (int)(i / M);
  int c = (int)(i - (long long)node * M);
  float dn = dinv[node];
  agg[i] = t[i] * (dn * dn) + b[c];
}

// agg[dst[e], :] += t[src[e], :] * en[e]   (edge-parallel scatter-add)
// One wave handles one full 128-float row (32 lanes x float4) -> fully
// coalesced 512B gathers; atomics land in L2 (feature matrix is 25.6MB << 192MB).
template <int M>
__global__ void k_edges(const int* __restrict__ src, const int* __restrict__ dst,
                        const float* __restrict__ en, const float* __restrict__ t,
                        float* __restrict__ agg, int E) {
  constexpr int G = M / 4;
  long long i = (long long)blockIdx.x * blockDim.x + threadIdx.x;
  long long total = (long long)E * G;
  if (i >= total) return;
  int e = (int)(i / G);
  int g = ((int)(i - (long long)e * G)) << 2;
  int s = src[e], d = dst[e];
  float nrm = en[e];
  const float4 v = *(const float4*)&t[(size_t)s * M + g];
  float* ap = &agg[(size_t)d * M + g];
  atom_add_f32(ap + 0, v.x * nrm);
  atom_add_f32(ap + 1, v.y * nrm);
  atom_add_f32(ap + 2, v.z * nrm);
  atom_add_f32(ap + 3, v.w * nrm);
}

// out[n] = relu(agg[n,:64]) . Wf + bf
__global__ void k_final(const float* __restrict__ agg, const float* __restrict__ Wf,
                        const float* __restrict__ bf, float* __restrict__ out, int N) {
  int n = blockIdx.x * blockDim.x + threadIdx.x;
  if (n >= N) return;
  const float4* row = (const float4*)&agg[(size_t)n * 64];
  float acc = bf[0];
#pragma unroll
  for (int j = 0; j < 16; ++j) {
    float4 h = row[j];
    float4 w = ((const float4*)Wf)[j];
    acc += fmaxf(h.x, 0.0f) * w.x + fmaxf(h.y, 0.0f) * w.y +
           fmaxf(h.z, 0.0f) * w.z + fmaxf(h.w, 0.0f) * w.w;
  }
  out[n] = acc;
}

extern "C" void kernel_launch(void* const* d_in, const int* in_sizes, int n_in,
                              void* d_out, int out_size, void* d_ws, size_t ws_size,
                              hipStream_t stream) {
  const float* x  = (const float*)d_in[0];
  const int*   src = (const int*)d_in[1];
  const int*   dst = (const int*)d_in[2];
  const float* W1 = (const float*)d_in[3];
  const float* b1 = (const float*)d_in[4];
  const float* W2 = (const float*)d_in[5];
  const float* b2 = (const float*)d_in[6];
  const float* W3 = (const float*)d_in[7];
  const float* b3 = (const float*)d_in[8];
  const float* Wf = (const float*)d_in[9];
  const float* bf = (const float*)d_in[10];
  float* out = (float*)d_out;

  const int IN = 128, HID = 128;
  const int N = in_sizes[0] / IN;   // 50000 (divisible by 16)
  const int E = in_sizes[1];        // 800000

  char* ws = (char*)d_ws;
  size_t off = 0;
  auto carve = [&](size_t bytes) -> void* {
    void* p = ws + off;
    off += (bytes + 255) & ~(size_t)255;
    return p;
  };
  float* dinv = (float*)carve((size_t)N * 4);
  float* deg  = (float*)carve((size_t)N * 4);
  float* en   = (float*)carve((size_t)E * 4);
  float* tbuf = (float*)carve((size_t)N * HID * 4);  // post-GEMM transform
  float* agg  = (float*)carve((size_t)N * HID * 4);  // aggregated output

  auto cdiv = [](long long a, long long b) { return (unsigned)((a + b - 1) / b); };

  // Degree + symmetric norms (computed once, reused for all 3 layers)
  k_zero<<<cdiv(N, 256), 256, 0, stream>>>(deg, N);
  k_deg<<<cdiv(E, 256), 256, 0, stream>>>(dst, deg, E);
  k_dinv<<<cdiv(N, 256), 256, 0, stream>>>(deg, dinv, N);
  k_edgenorm<<<cdiv(E, 256), 256, 0, stream>>>(src, dst, dinv, en, E);

  const unsigned gemmGrid = cdiv(N, 16);

  // Layer 1: t = x @ W1 ; agg = scatter(t) + t*selfnorm + b1
  k_gemm_wmma<128, 128, false><<<gemmGrid, 256, 0, stream>>>(x, W1, tbuf, N);
  k_self<128><<<cdiv((long long)N * 128, 256), 256, 0, stream>>>(tbuf, dinv, b1, agg, N);
  k_edges<128><<<cdiv((long long)E * 32, 256), 256, 0, stream>>>(src, dst, en, tbuf, agg, E);

  // Layer 2 (ReLU of layer-1 output fused into GEMM A-load)
  k_gemm_wmma<128, 128, true><<<gemmGrid, 256, 0, stream>>>(agg, W2, tbuf, N);
  k_self<128><<<cdiv((long long)N * 128, 256), 256, 0, stream>>>(tbuf, dinv, b2, agg, N);
  k_edges<128><<<cdiv((long long)E * 32, 256), 256, 0, stream>>>(src, dst, en, tbuf, agg, E);

  // Layer 3: 128 -> 64
  k_gemm_wmma<128, 64, true><<<gemmGrid, 128, 0, stream>>>(agg, W3, tbuf, N);
  k_self<64><<<cdiv((long long)N * 64, 256), 256, 0, stream>>>(tbuf, dinv, b3, agg, N);
  k_edges<64><<<cdiv((long long)E * 16, 256), 256, 0, stream>>>(src, dst, en, tbuf, agg, E);

  // Final linear head: out = relu(agg) @ Wf + bf
  k_final<<<cdiv(N, 256), 256, 0, stream>>>(agg, Wf, bf, out, N);
}